// PointNetLK_mod_76544907149593
// MI455X (gfx1250) — compile-verified
//
#include <hip/hip_runtime.h>
#include <math.h>

typedef __attribute__((ext_vector_type(2))) float v2f;
typedef __attribute__((ext_vector_type(8))) float v8f;
typedef __attribute__((ext_vector_type(4))) unsigned int u32x4;
typedef __attribute__((ext_vector_type(8))) int i32x8;
typedef __attribute__((ext_vector_type(4))) int i32x4;

#define B_     8
#define N_     2048
#define KSEL   512
#define FDIM   1024
#define DELTA_ 0.01f
#define XTOL_  1e-7f

// ---------------- device helpers ----------------

__device__ inline void se3_exp_dev(const float* xi, float* T) {
  float wx = xi[0], wy = xi[1], wz = xi[2];
  float vx = xi[3], vy = xi[4], vz = xi[5];
  float th2  = wx*wx + wy*wy + wz*wz;
  float th   = sqrtf(th2);
  float safe = fmaxf(th, 1e-8f);
  float A, Bc, C;
  if (th < 1e-4f) {
    A  = 1.f - th2 * (1.f/6.f);
    Bc = 0.5f - th2 * (1.f/24.f);
    C  = (1.f/6.f) - th2 * (1.f/120.f);
  } else {
    float s = sinf(safe), c = cosf(safe);
    A  = s / safe;
    Bc = (1.f - c) / (safe*safe);
    C  = (safe - s) / (safe*safe*safe);
  }
  float W[9]  = {0.f,-wz, wy,  wz, 0.f,-wx,  -wy, wx, 0.f};
  float W2[9];
  for (int i = 0; i < 3; ++i)
    for (int j = 0; j < 3; ++j) {
      float s = 0.f;
      for (int k = 0; k < 3; ++k) s += W[i*3+k] * W[k*3+j];
      W2[i*3+j] = s;
    }
  float R[9], V[9];
  for (int i = 0; i < 9; ++i) {
    float id = (i % 4 == 0) ? 1.f : 0.f;
    R[i] = id + A  * W[i] + Bc * W2[i];
    V[i] = id + Bc * W[i] + C  * W2[i];
  }
  float t0 = V[0]*vx + V[1]*vy + V[2]*vz;
  float t1 = V[3]*vx + V[4]*vy + V[5]*vz;
  float t2 = V[6]*vx + V[7]*vy + V[8]*vz;
  T[0]=R[0];  T[1]=R[1];  T[2]=R[2];   T[3]=t0;
  T[4]=R[3];  T[5]=R[4];  T[6]=R[5];   T[7]=t1;
  T[8]=R[6];  T[9]=R[7];  T[10]=R[8];  T[11]=t2;
  T[12]=0.f;  T[13]=0.f;  T[14]=0.f;   T[15]=1.f;
}

__device__ inline void invert6(const float* H, float* Hi) {
  float M[6][12];
  for (int i = 0; i < 6; ++i)
    for (int j = 0; j < 6; ++j) {
      M[i][j]   = H[i*6+j];
      M[i][6+j] = (i == j) ? 1.f : 0.f;
    }
  for (int c = 0; c < 6; ++c) {
    int piv = c; float best = fabsf(M[c][c]);
    for (int r = c+1; r < 6; ++r) {
      float a = fabsf(M[r][c]);
      if (a > best) { best = a; piv = r; }
    }
    if (piv != c)
      for (int j = 0; j < 12; ++j) { float t = M[c][j]; M[c][j] = M[piv][j]; M[piv][j] = t; }
    float inv = 1.f / M[c][c];
    for (int j = 0; j < 12; ++j) M[c][j] *= inv;
    for (int r = 0; r < 6; ++r) if (r != c) {
      float f = M[r][c];
      for (int j = 0; j < 12; ++j) M[r][j] -= f * M[c][j];
    }
  }
  for (int i = 0; i < 6; ++i)
    for (int j = 0; j < 6; ++j) Hi[i*6+j] = M[i][6+j];
}

// ---- TDM: stage one 128(K) x 16(cols) f32 tile of W5 into LDS ----
// D# per CDNA5 ISA ch.8: count=1, type=2 ("image"), data_size=4B,
// tensor_dim0=1024 (W5 row width), tensor_dim0_stride=1024,
// tile_dim0=16, tile_dim1=128, LDS padding 4 dwords every 16 dwords
// => LDS row stride = 20 floats (bank-conflict-free for 16-lane column reads).

#define W5_STR 20

__device__ inline void tdm_load_w5_tile(const float* gsrc, float* ldsdst) {
  unsigned long long ga = (unsigned long long)(size_t)gsrc;
  unsigned lds = (unsigned)(size_t)(void*)ldsdst;   // flat LDS ptr: low 32 bits = LDS byte address
  u32x4 g0;
  g0[0] = 1u;                                       // count=1, user descriptor
  g0[1] = lds;                                      // lds_addr (bytes)
  g0[2] = (unsigned)(ga & 0xffffffffu);             // global_addr[31:0]   (bits 95:64)
  g0[3] = (unsigned)((ga >> 32) & 0x01ffffffu)      // global_addr[56:32]  (bits 120:96)
        | (2u << 30);                               // type=2              (bits 127:126)
  i32x8 g1;
  g1[0] = (int)((2u << 16)                          // data_size = 4 bytes
              | (1u << 20)                          // pad_enable
              | (3u << 22)                          // pad_interval: 16 dwords
              | (3u << 25));                        // pad_amount: 4 dwords
  g1[1] = (int)(1024u << 16);                       // tensor_dim0[15:0] @ bits 63:48
  g1[2] = (int)(128u << 16);                        // tensor_dim0[31:16]=0 | tensor_dim1[15:0]=128
  g1[3] = (int)(16u << 16);                         // tensor_dim1[31:16]=0 | tile_dim0=16
  g1[4] = (int)(128u);                              // tile_dim1=128 | tile_dim2=0
  g1[5] = (int)(1024);                              // tensor_dim0_stride[31:0]
  g1[6] = 0;                                        // stride0[47:32] | tensor_dim1_stride[15:0]
  g1[7] = 0;
  i32x4 z4 = {0, 0, 0, 0};
  i32x8 z8 = {0, 0, 0, 0, 0, 0, 0, 0};
  // amdgpu-toolchain (clang-23) 6-arg form: (g0, g1, g2, g3, extra, cpol)
  __builtin_amdgcn_tensor_load_to_lds(g0, g1, z4, z4, z8, 0);
}

// ---------------- means / centering ----------------

__global__ void means_kernel(const float* __restrict__ tpl, const float* __restrict__ src,
                             float* __restrict__ tm, float* __restrict__ sm) {
  __shared__ float red[256];
  int b = blockIdx.x, tid = threadIdx.x;
  for (int which = 0; which < 2; ++which) {
    const float* p = which ? src : tpl;
    float* o = which ? sm : tm;
    for (int d = 0; d < 3; ++d) {
      float s = 0.f;
      for (int n = tid; n < N_; n += 256) s += p[((size_t)b*N_ + n)*3 + d];
      red[tid] = s; __syncthreads();
      for (int off = 128; off > 0; off >>= 1) {
        if (tid < off) red[tid] += red[tid+off];
        __syncthreads();
      }
      if (tid == 0) o[b*3+d] = red[0] * (1.f / N_);
      __syncthreads();
    }
  }
}

__global__ void center_kernel(const float* __restrict__ tpl, const float* __restrict__ src,
                              const float* __restrict__ tm, const float* __restrict__ sm,
                              float* __restrict__ tc, float* __restrict__ sc) {
  int i = blockIdx.x * blockDim.x + threadIdx.x;
  if (i >= B_ * N_) return;
  int b = i / N_;
  for (int d = 0; d < 3; ++d) {
    tc[(size_t)i*3+d] = tpl[(size_t)i*3+d] - tm[b*3+d];
    sc[(size_t)i*3+d] = src[(size_t)i*3+d] - sm[b*3+d];
  }
}

// ---------------- init ----------------

__global__ void init_kernel(const float* __restrict__ W1,
                            float* __restrict__ W1p, float* __restrict__ Dall,
                            float* __restrict__ est_T, int* __restrict__ done,
                            float* __restrict__ f0, float* __restrict__ fpert) {
  int tid = threadIdx.x;
  if (tid < 48) {
    float xi[6] = {0.f,0.f,0.f,0.f,0.f,0.f};
    xi[tid % 6] = -DELTA_;
    se3_exp_dev(xi, Dall + tid*16);
  }
  if (tid < 8) {
    float* T = est_T + tid*16;
    for (int i = 0; i < 16; ++i) T[i] = (i % 5 == 0) ? 1.f : 0.f;
  }
  if (tid == 0) *done = 0;
  for (int i = tid; i < 4*64; i += 256) {
    int r = i >> 6, c = i & 63;
    W1p[i] = (r < 3) ? W1[r*64 + c] : 0.f;
  }
  for (int i = tid; i < B_*FDIM; i += 256)   f0[i]    = 0.f;
  for (int i = tid; i < B_*6*FDIM; i += 256) fpert[i] = 0.f;
}

__global__ void zero_kernel(float* __restrict__ p, int n) {
  int i = blockIdx.x * blockDim.x + threadIdx.x;
  if (i < n) p[i] = 0.f;
}

// ---------------- fused PointNet feature kernel ----------------
// 128 threads = 4 waves; each wave owns one 16-point WMMA row tile through
// all 5 layers (LDS ping-pong, wave-private rows => no barriers in layers 1-4).
// Layer 5 (128->1024, 87% of FLOPs): weight tiles staged by the Tensor Data
// Mover into double-buffered LDS (one tensor_load_to_lds per 128x16 tile,
// issued by wave 0, s_wait_tensorcnt + one barrier per tile), WMMAs of tile t
// overlap the TDM of tile t+1; fused ReLU + max-pool via uint atomicMax.

#define TILE_PTS   64
#define FEAT_BLOCK 128
#define STR_A      68
#define STR_B      132

__global__ __launch_bounds__(FEAT_BLOCK)
void feat_kernel(const float* __restrict__ pts, int P, int group_div,
                 const float* __restrict__ transforms,
                 const float* __restrict__ W1p, const float* __restrict__ b1,
                 const float* __restrict__ W2,  const float* __restrict__ b2,
                 const float* __restrict__ W3,  const float* __restrict__ b3,
                 const float* __restrict__ W4,  const float* __restrict__ b4,
                 const float* __restrict__ W5,  const float* __restrict__ b5,
                 float* __restrict__ feat_out, int blocks_per_group) {
  __shared__ float sT[16];
  __shared__ float Xin[TILE_PTS * 4];
  __shared__ float A1[TILE_PTS * STR_A];
  __shared__ float A2[TILE_PTS * STR_B];
  __shared__ float Wt[2][128 * W5_STR];   // TDM-staged W5 tiles, double buffered

  const int tid  = threadIdx.x;
  const int g    = blockIdx.x / blocks_per_group;
  const int tile = blockIdx.x - g * blocks_per_group;
  const int p0   = tile * TILE_PTS;
  const float* src = pts + (size_t)(g / group_div) * (size_t)P * 3;

  if (tid < 16) sT[tid] = transforms[g*16 + tid];
  __syncthreads();

  for (int i = tid; i < TILE_PTS; i += FEAT_BLOCK) {
    int p = p0 + i;
    float x = 0.f, y = 0.f, z = 0.f;
    if (p < P) { x = src[(size_t)p*3+0]; y = src[(size_t)p*3+1]; z = src[(size_t)p*3+2]; }
    float tx = sT[0]*x + sT[1]*y + sT[2]*z  + sT[3];
    float ty = sT[4]*x + sT[5]*y + sT[6]*z  + sT[7];
    float tz = sT[8]*x + sT[9]*y + sT[10]*z + sT[11];
    Xin[i*4+0] = tx; Xin[i*4+1] = ty; Xin[i*4+2] = tz; Xin[i*4+3] = 0.f;
  }
  __syncthreads();

  const int lane = tid & 31;
  const int wave = tid >> 5;
  const int half = lane >> 4;   // lane half selects K-pair {0,1} vs {2,3}
  const int nl   = lane & 15;
  const int rowb = wave * 16;

  // kick off TDM staging of the first W5 tile; it hides behind layers 1-4
  if (wave == 0) tdm_load_w5_tile(W5, &Wt[0][0]);

  auto layer = [&](const float* X, int xstr, const float* Wg, const float* bg,
                   float* Y, int ystr, int K, int Nout) {
    for (int c0 = 0; c0 < Nout; c0 += 16) {
      float bias = bg[c0 + nl];
      v8f acc;
#pragma unroll
      for (int r = 0; r < 8; ++r) acc[r] = bias;
      for (int k0 = 0; k0 < K; k0 += 4) {
        int ka = k0 + half*2;
        v2f a, bm;
        const float* xr = X + (rowb + nl)*xstr + ka;
        a.x  = xr[0];
        a.y  = xr[1];
        bm.x = Wg[(size_t)ka*Nout     + c0 + nl];
        bm.y = Wg[(size_t)(ka+1)*Nout + c0 + nl];
        if (k0 + 8 < K)
          __builtin_prefetch(Wg + (size_t)(ka+8)*Nout + c0 + nl, 0, 0);
        acc = __builtin_amdgcn_wmma_f32_16x16x4_f32(false, a, false, bm,
                                                    (short)0, acc, false, false);
      }
#pragma unroll
      for (int r = 0; r < 8; ++r) {
        float v = fmaxf(acc[r], 0.f);  // ReLU
        Y[(rowb + half*8 + r)*ystr + c0 + nl] = v;
      }
    }
  };

  layer(Xin, 4,     W1p, b1, A1, STR_A, 4,   64);
  layer(A1,  STR_A, W2,  b2, A2, STR_B, 64,  64);
  layer(A2,  STR_B, W3,  b3, A1, STR_A, 64,  64);
  layer(A1,  STR_A, W4,  b4, A2, STR_B, 64, 128);

  // Layer 5, TDM double-buffered + streamed max-pool
  float* fout = feat_out + (size_t)g * FDIM;
  for (int t = 0; t < FDIM/16; ++t) {
    const int cur = t & 1;
    if (wave == 0) __builtin_amdgcn_s_wait_tensorcnt(0);
    __syncthreads();  // tile t staged; all waves done reading buf[1-cur] (tile t-1)
    if (wave == 0 && t + 1 < FDIM/16)
      tdm_load_w5_tile(W5 + (size_t)(t+1)*16, &Wt[1-cur][0]);

    const int c0 = t * 16;
    float bias = b5[c0 + nl];
    v8f acc;
#pragma unroll
    for (int r = 0; r < 8; ++r) acc[r] = bias;
    const float* Wl = &Wt[cur][0];
    for (int k0 = 0; k0 < 128; k0 += 4) {
      int ka = k0 + half*2;
      v2f a, bm;
      const float* xr = A2 + (rowb + nl)*STR_B + ka;
      a.x  = xr[0];
      a.y  = xr[1];
      bm.x = Wl[ka*W5_STR     + nl];
      bm.y = Wl[(ka+1)*W5_STR + nl];
      acc = __builtin_amdgcn_wmma_f32_16x16x4_f32(false, a, false, bm,
                                                  (short)0, acc, false, false);
    }
    float m = 0.f;                       // ReLU floor doubles as max identity
#pragma unroll
    for (int r = 0; r < 8; ++r) m = fmaxf(m, acc[r]);
    m = fmaxf(m, __shfl_xor(m, 16, 32));
    if (half == 0)
      atomicMax(reinterpret_cast<unsigned int*>(&fout[c0 + nl]), __float_as_uint(m));
  }
}

// ---------------- brute-force NN min distance ----------------

__global__ void mind_kernel(const float* __restrict__ tc, const float* __restrict__ sc,
                            float* __restrict__ mind) {
  __shared__ float S[N_ * 3];
  int b = blockIdx.x >> 3;
  int chunk = blockIdx.x & 7;
  int tid = threadIdx.x;
  for (int i = tid; i < N_*3; i += 256) S[i] = sc[(size_t)b*N_*3 + i];
  __syncthreads();
  int n = chunk*256 + tid;
  float px = tc[((size_t)b*N_+n)*3+0];
  float py = tc[((size_t)b*N_+n)*3+1];
  float pz = tc[((size_t)b*N_+n)*3+2];
  float best = 3.4e38f;
  for (int m = 0; m < N_; ++m) {
    float dx = px - S[m*3+0], dy = py - S[m*3+1], dz = pz - S[m*3+2];
    best = fminf(best, dx*dx + dy*dy + dz*dz);
  }
  mind[(size_t)b*N_ + n] = best;
}

// ---------------- top-K via bitonic sort ----------------

__global__ void topk_kernel(const float* __restrict__ mind, const float* __restrict__ tc,
                            float* __restrict__ topk_pts) {
  __shared__ unsigned long long keys[N_];
  int b = blockIdx.x, tid = threadIdx.x;   // 1024 threads
  for (int i = tid; i < N_; i += 1024) {
    unsigned fb = __float_as_uint(mind[(size_t)b*N_ + i]);  // mind >= 0 => order-preserving
    keys[i] = ((unsigned long long)fb << 32) | (unsigned)i;
  }
  __syncthreads();
  for (int k = 2; k <= N_; k <<= 1)
    for (int j = k >> 1; j > 0; j >>= 1) {
      for (int i = tid; i < N_; i += 1024) {
        int ixj = i ^ j;
        if (ixj > i) {
          bool up = ((i & k) == 0);
          unsigned long long a = keys[i], c = keys[ixj];
          if ((a > c) == up) { keys[i] = c; keys[ixj] = a; }
        }
      }
      __syncthreads();
    }
  for (int t = tid; t < KSEL; t += 1024) {
    int idx = (int)(keys[t] & 0xffffffffu);
    for (int d = 0; d < 3; ++d)
      topk_pts[((size_t)b*KSEL + t)*3 + d] = tc[((size_t)b*N_ + idx)*3 + d];
  }
}

// ---------------- Jacobian, H = J^T J, pinv = H^-1 J^T ----------------

__global__ void jac_kernel(const float* __restrict__ f0, const float* __restrict__ fpert,
                           float* __restrict__ J, float* __restrict__ pinv) {
  __shared__ float red[256];
  __shared__ float H[36], Hi[36];
  int b = blockIdx.x, tid = threadIdx.x;
  for (int f = tid; f < FDIM; f += 256)
    for (int i = 0; i < 6; ++i)
      J[((size_t)b*FDIM + f)*6 + i] =
          (f0[(size_t)b*FDIM + f] - fpert[((size_t)(b*6+i))*FDIM + f]) * (1.f / DELTA_);
  __syncthreads();
  for (int p = 0; p < 36; ++p) {
    int i = p / 6, j = p % 6;
    float s = 0.f;
    for (int f = tid; f < FDIM; f += 256)
      s += J[((size_t)b*FDIM + f)*6 + i] * J[((size_t)b*FDIM + f)*6 + j];
    red[tid] = s; __syncthreads();
    for (int off = 128; off > 0; off >>= 1) {
      if (tid < off) red[tid] += red[tid+off];
      __syncthreads();
    }
    if (tid == 0) H[p] = red[0];
    __syncthreads();
  }
  if (tid == 0) invert6(H, Hi);
  __syncthreads();
  for (int f = tid; f < FDIM; f += 256)
    for (int i = 0; i < 6; ++i) {
      float s = 0.f;
      for (int j = 0; j < 6; ++j) s += Hi[i*6+j] * J[((size_t)b*FDIM + f)*6 + j];
      pinv[(size_t)b*6*FDIM + (size_t)i*FDIM + f] = s;
    }
}

// ---------------- one Gauss-Newton step ----------------

__global__ void step_kernel(const float* __restrict__ pinv, const float* __restrict__ f0,
                            const float* __restrict__ feat, float* __restrict__ est_T,
                            int* __restrict__ done, const int* __restrict__ maxiter, int it) {
  if (it >= *maxiter) return;
  if (*done) return;                       // scalar done: all batches freeze together
  int tid = threadIdx.x;                   // 256 = 8 waves
  int wave = tid >> 5, lane = tid & 31;
  __shared__ float pose[48];
  for (int p = wave; p < 48; p += 8) {
    int b = p / 6, i = p % 6;
    float s = 0.f;
    for (int f = lane; f < FDIM; f += 32)
      s += pinv[(size_t)b*6*FDIM + (size_t)i*FDIM + f] *
           (feat[(size_t)b*FDIM + f] - f0[(size_t)b*FDIM + f]);
    for (int off = 16; off > 0; off >>= 1) s += __shfl_down(s, off, 32);
    if (lane == 0) pose[p] = -s;
  }
  __syncthreads();
  if (tid == 0) {
    float check = 0.f;
    for (int b = 0; b < 8; ++b) {
      float n2 = 0.f;
      for (int i = 0; i < 6; ++i) n2 += pose[b*6+i] * pose[b*6+i];
      check = fmaxf(check, sqrtf(n2));
    }
    if (check < XTOL_) {
      *done = 1;
    } else {
      for (int b = 0; b < 8; ++b) {
        float dT[16], nT[16];
        se3_exp_dev(&pose[b*6], dT);
        const float* E = est_T + b*16;
        for (int i = 0; i < 4; ++i)
          for (int j = 0; j < 4; ++j) {
            float s = 0.f;
            for (int k = 0; k < 4; ++k) s += dT[i*4+k] * E[k*4+j];
            nT[i*4+j] = s;
          }
        for (int i = 0; i < 16; ++i) est_T[b*16+i] = nT[i];
      }
    }
  }
}

// ---------------- final transform of the ORIGINAL source cloud ----------------

__global__ void final_kernel(const float* __restrict__ src, const float* __restrict__ est_T,
                             const float* __restrict__ tm, const float* __restrict__ sm,
                             float* __restrict__ out) {
  int idx = blockIdx.x * blockDim.x + threadIdx.x;
  if (idx >= B_ * N_) return;
  int b = idx >> 11;
  const float* T = est_T + b*16;
  float R0=T[0], R1=T[1], R2=T[2], R3=T[4], R4=T[5], R5=T[6], R6=T[8], R7=T[9], R8=T[10];
  float sx = sm[b*3+0], sy = sm[b*3+1], sz = sm[b*3+2];
  float t0 = T[3]  + tm[b*3+0] - (R0*sx + R1*sy + R2*sz);
  float t1 = T[7]  + tm[b*3+1] - (R3*sx + R4*sy + R5*sz);
  float t2 = T[11] + tm[b*3+2] - (R6*sx + R7*sy + R8*sz);
  float px = src[(size_t)idx*3+0], py = src[(size_t)idx*3+1], pz = src[(size_t)idx*3+2];
  out[(size_t)idx*3+0] = R0*px + R1*py + R2*pz + t0;
  out[(size_t)idx*3+1] = R3*px + R4*py + R5*pz + t1;
  out[(size_t)idx*3+2] = R6*px + R7*py + R8*pz + t2;
}

// ---------------- host-side orchestration ----------------

extern "C" void kernel_launch(void* const* d_in, const int* in_sizes, int n_in,
                              void* d_out, int out_size, void* d_ws, size_t ws_size,
                              hipStream_t stream) {
  (void)in_sizes; (void)n_in; (void)out_size; (void)ws_size;
  const float* W1 = (const float*)d_in[0];
  const float* b1 = (const float*)d_in[1];
  const float* W2 = (const float*)d_in[2];
  const float* b2 = (const float*)d_in[3];
  const float* W3 = (const float*)d_in[4];
  const float* b3 = (const float*)d_in[5];
  const float* W4 = (const float*)d_in[6];
  const float* b4 = (const float*)d_in[7];
  const float* W5 = (const float*)d_in[8];
  const float* b5 = (const float*)d_in[9];
  const float* tpl = (const float*)d_in[10];
  const float* src = (const float*)d_in[11];
  const int* maxiter = (const int*)d_in[12];
  float* out = (float*)d_out;

  char* w = (char*)d_ws;
  auto alloc = [&](size_t nbytes) -> void* {
    void* p = (void*)w;
    w += (nbytes + 255) & ~(size_t)255;
    return p;
  };
  float* tm    = (float*)alloc(B_*3*sizeof(float));
  float* sm    = (float*)alloc(B_*3*sizeof(float));
  float* tc    = (float*)alloc((size_t)B_*N_*3*sizeof(float));
  float* sc    = (float*)alloc((size_t)B_*N_*3*sizeof(float));
  float* f0    = (float*)alloc((size_t)B_*FDIM*sizeof(float));
  float* mind  = (float*)alloc((size_t)B_*N_*sizeof(float));
  float* tpk   = (float*)alloc((size_t)B_*KSEL*3*sizeof(float));
  float* W1p   = (float*)alloc(4*64*sizeof(float));
  float* Dall  = (float*)alloc(48*16*sizeof(float));
  float* fpert = (float*)alloc((size_t)48*FDIM*sizeof(float));
  float* J     = (float*)alloc((size_t)B_*FDIM*6*sizeof(float));
  float* pinv  = (float*)alloc((size_t)B_*6*FDIM*sizeof(float));
  float* estT  = (float*)alloc(B_*16*sizeof(float));
  float* fit   = (float*)alloc((size_t)B_*FDIM*sizeof(float));
  int*   done  = (int*)alloc(16);

  means_kernel<<<B_, 256, 0, stream>>>(tpl, src, tm, sm);
  center_kernel<<<(B_*N_ + 255)/256, 256, 0, stream>>>(tpl, src, tm, sm, tc, sc);
  init_kernel<<<1, 256, 0, stream>>>(W1, W1p, Dall, estT, done, f0, fpert);

  feat_kernel<<<B_*(N_/TILE_PTS), FEAT_BLOCK, 0, stream>>>(
      tc, N_, 1, estT, W1p, b1, W2, b2, W3, b3, W4, b4, W5, b5, f0, N_/TILE_PTS);

  mind_kernel<<<B_*8, 256, 0, stream>>>(tc, sc, mind);
  topk_kernel<<<B_, 1024, 0, stream>>>(mind, tc, tpk);

  feat_kernel<<<48*(KSEL/TILE_PTS), FEAT_BLOCK, 0, stream>>>(
      tpk, KSEL, 6, Dall, W1p, b1, W2, b2, W3, b3, W4, b4, W5, b5, fpert, KSEL/TILE_PTS);

  jac_kernel<<<B_, 256, 0, stream>>>(f0, fpert, J, pinv);

  for (int it = 0; it < 10; ++it) {
    zero_kernel<<<(B_*FDIM + 255)/256, 256, 0, stream>>>(fit, B_*FDIM);
    feat_kernel<<<B_*(N_/TILE_PTS), FEAT_BLOCK, 0, stream>>>(
        sc, N_, 1, estT, W1p, b1, W2, b2, W3, b3, W4, b4, W5, b5, fit, N_/TILE_PTS);
    step_kernel<<<1, 256, 0, stream>>>(pinv, f0, fit, estT, done, maxiter, it);
  }

  final_kernel<<<(B_*N_ + 255)/256, 256, 0, stream>>>(src, estT, tm, sm, out);
}